// GCNNet_NC_8263517077505
// MI455X (gfx1250) — compile-verified
//
#include <hip/hip_runtime.h>
#include <hip/hip_bf16.h>

typedef __attribute__((ext_vector_type(2))) float v2f;
typedef __attribute__((ext_vector_type(8))) float v8f;

#define N_NODES 100000
#define N_EDGES 1600000
#define HID 128

// ---------------------------------------------------------------------------
// Degree / normalization kernels
// ---------------------------------------------------------------------------
__global__ void k_fill1(float* __restrict__ p, int n) {
    int i = blockIdx.x * blockDim.x + threadIdx.x;
    if (i < n) p[i] = 1.0f;  // self-loop contributes 1 to every degree
}

__global__ void k_deg_edges(const int* __restrict__ dst, float* __restrict__ deg, int e) {
    int i = blockIdx.x * blockDim.x + threadIdx.x;
    if (i < e) atomicAdd(&deg[dst[i]], 1.0f);
}

__global__ void k_rsqrt_inplace(float* __restrict__ p, int n) {
    int i = blockIdx.x * blockDim.x + threadIdx.x;
    if (i < n) p[i] = rsqrtf(p[i]);  // deg >= 1 always (self loop)
}

// ---------------------------------------------------------------------------
// FP32 WMMA GEMM: C[N x 128] = A[N x K] (row stride lda) * W[K x 128]
// One wave computes a 16x128 tile via 8 accumulators of v_wmma_f32_16x16x4_f32.
// N must be a multiple of 16 (100000 is), K a multiple of 4, F fixed = 128.
// ---------------------------------------------------------------------------
__global__ __launch_bounds__(256) void k_gemm_f128(
    const float* __restrict__ A, int lda,
    const float* __restrict__ W,
    float* __restrict__ C, int n, int k_dim) {
    const int wave = blockIdx.x * (blockDim.x >> 5) + (threadIdx.x >> 5);
    const int lane = threadIdx.x & 31;
    const int row0 = wave * 16;
    if (row0 >= n) return;  // wave-uniform: EXEC stays all-ones for WMMA

    const int half = lane >> 4;   // 0: K=k..k+1, 1: K=k+2..k+3
    const int l15  = lane & 15;
    const float* arow = A + (size_t)(row0 + l15) * (size_t)lda;

    v8f acc[8] = {};

    for (int k = 0; k < k_dim; k += 4) {
        const int ka = k + half * 2;
        v2f a;
        a.x = arow[ka];
        a.y = arow[ka + 1];
        const float* wr0 = W + (size_t)ka * 128 + l15;
        const float* wr1 = wr0 + 128;
#pragma unroll
        for (int j = 0; j < 8; ++j) {
            v2f b;
            b.x = wr0[j * 16];
            b.y = wr1[j * 16];
            acc[j] = __builtin_amdgcn_wmma_f32_16x16x4_f32(
                false, a, false, b, (short)0, acc[j], false, false);
        }
    }

#pragma unroll
    for (int j = 0; j < 8; ++j) {
        const int col = j * 16 + l15;
#pragma unroll
        for (int g = 0; g < 8; ++g) {
            const int m = g + half * 8;
            C[(size_t)(row0 + m) * 128 + col] = acc[j][g];
        }
    }
}

// ---------------------------------------------------------------------------
// agg[n][c] = dinv[n]^2 * t[n][c]   (self-loop term of the sym-normalized sum)
// ---------------------------------------------------------------------------
__global__ void k_selfloop(const float* __restrict__ t, const float* __restrict__ dinv,
                           float* __restrict__ agg, int total) {
    int i = blockIdx.x * blockDim.x + threadIdx.x;
    if (i >= total) return;
    int node = i >> 7;
    float dv = dinv[node];
    agg[i] = dv * dv * t[i];
}

// ---------------------------------------------------------------------------
// Edge scatter: one wave per edge; lane handles 4 contiguous feature columns.
// agg[dst] += dinv[src]*dinv[dst] * t[src]   -> contiguous 512B gather + atomics
// ---------------------------------------------------------------------------
__global__ void k_scatter(const int* __restrict__ src, const int* __restrict__ dst,
                          const float* __restrict__ dinv, const float* __restrict__ t,
                          float* __restrict__ agg, int e) {
    long long tid = (long long)blockIdx.x * blockDim.x + threadIdx.x;
    int edge = (int)(tid >> 5);
    if (edge >= e) return;
    int lane = (int)(tid & 31);
    int s = src[edge], d = dst[edge];
    float w = dinv[s] * dinv[d];
    const float4 v = *reinterpret_cast<const float4*>(t + (size_t)s * 128 + lane * 4);
    float* o = agg + (size_t)d * 128 + lane * 4;
    atomicAdd(o + 0, w * v.x);
    atomicAdd(o + 1, w * v.y);
    atomicAdd(o + 2, w * v.z);
    atomicAdd(o + 3, w * v.w);
}

// ---------------------------------------------------------------------------
// emb[n][col_off + c] = relu(agg[n][c] + b[c])   (emb row stride = 384)
// ---------------------------------------------------------------------------
__global__ void k_bias_relu_emb(const float* __restrict__ agg, const float* __restrict__ b,
                                float* __restrict__ emb, int col_off, int total) {
    int i = blockIdx.x * blockDim.x + threadIdx.x;
    if (i >= total) return;
    int node = i >> 7;
    int c = i & 127;
    emb[(size_t)node * 384 + col_off + c] = fmaxf(agg[i] + b[c], 0.0f);
}

// ---------------------------------------------------------------------------
// t[i] = elu(t[i] + b[c])
// ---------------------------------------------------------------------------
__global__ void k_bias_elu(float* __restrict__ t, const float* __restrict__ b, int total) {
    int i = blockIdx.x * blockDim.x + threadIdx.x;
    if (i >= total) return;
    float v = t[i] + b[i & 127];
    t[i] = v > 0.0f ? v : (expf(v) - 1.0f);
}

// ---------------------------------------------------------------------------
// Head: logits = t @ Wm1 + bm1 ; probs = softmax(logits).
// Wm1 (128x40) + bm1 staged in LDS; one thread per node.
// ---------------------------------------------------------------------------
__global__ void k_head(const float* __restrict__ t, const float* __restrict__ Wm1,
                       const float* __restrict__ bm1, float* __restrict__ logits,
                       float* __restrict__ probs, int n) {
    __shared__ float Ws[128 * 40];
    __shared__ float bs[40];
    for (int i = threadIdx.x; i < 128 * 40; i += blockDim.x) Ws[i] = Wm1[i];
    if (threadIdx.x < 40) bs[threadIdx.x] = bm1[threadIdx.x];
    __syncthreads();

    int node = blockIdx.x * blockDim.x + threadIdx.x;
    if (node >= n) return;

    float acc[40];
#pragma unroll
    for (int c = 0; c < 40; ++c) acc[c] = bs[c];

    const float* tr = t + (size_t)node * 128;
    for (int k = 0; k < 128; ++k) {
        float tv = tr[k];
        const float* wr = &Ws[k * 40];
#pragma unroll
        for (int c = 0; c < 40; ++c) acc[c] = fmaf(tv, wr[c], acc[c]);
    }

    float* lo = logits + (size_t)node * 40;
#pragma unroll
    for (int c = 0; c < 40; ++c) lo[c] = acc[c];

    float m = acc[0];
#pragma unroll
    for (int c = 1; c < 40; ++c) m = fmaxf(m, acc[c]);
    float ssum = 0.0f;
#pragma unroll
    for (int c = 0; c < 40; ++c) { acc[c] = expf(acc[c] - m); ssum += acc[c]; }
    float inv = 1.0f / ssum;
    float* pr = probs + (size_t)node * 40;
#pragma unroll
    for (int c = 0; c < 40; ++c) pr[c] = acc[c] * inv;
}

// ---------------------------------------------------------------------------
// Orchestration
// ---------------------------------------------------------------------------
extern "C" void kernel_launch(void* const* d_in, const int* in_sizes, int n_in,
                              void* d_out, int out_size, void* d_ws, size_t ws_size,
                              hipStream_t stream) {
    (void)in_sizes; (void)n_in; (void)out_size; (void)ws_size;

    const float* x   = (const float*)d_in[0];
    const int*   ei  = (const int*)d_in[1];
    const float* W0  = (const float*)d_in[2];
    const float* b0  = (const float*)d_in[3];
    const float* W1  = (const float*)d_in[4];
    const float* b1  = (const float*)d_in[5];
    const float* W2  = (const float*)d_in[6];
    const float* b2  = (const float*)d_in[7];
    const float* Wm0 = (const float*)d_in[8];
    const float* bm0 = (const float*)d_in[9];
    const float* Wm1 = (const float*)d_in[10];
    const float* bm1 = (const float*)d_in[11];

    const int N = N_NODES, E = N_EDGES;
    const int* src = ei;
    const int* dst = ei + E;

    // workspace layout (floats): dinv | t (N*128) | agg (N*128)  ~= 103 MB
    float* ws   = (float*)d_ws;
    float* dinv = ws;
    float* t    = ws + (1 << 17);
    float* agg  = t + (size_t)N * 128;

    // output layout (floats): logits (N*40) | probs (N*40) | emb (N*384)
    float* logits = (float*)d_out;
    float* probs  = logits + (size_t)N * 40;
    float* emb    = probs + (size_t)N * 40;

    const int TPB = 256;
    const int nf  = N * 128;                      // 12.8M
    const int gridNF = (nf + TPB - 1) / TPB;
    const int gemm_blocks = ((N / 16) + 7) / 8;   // 8 waves (16-row tiles) per block
    const long long scat_threads = (long long)E * 32;
    const int gridScat = (int)((scat_threads + TPB - 1) / TPB);

    // --- symmetric normalization coefficients ---
    k_fill1<<<(N + TPB - 1) / TPB, TPB, 0, stream>>>(dinv, N);
    k_deg_edges<<<(E + TPB - 1) / TPB, TPB, 0, stream>>>(dst, dinv, E);
    k_rsqrt_inplace<<<(N + TPB - 1) / TPB, TPB, 0, stream>>>(dinv, N);

    // --- GCN layer 0: x[N,512] @ W0 ---
    k_gemm_f128<<<gemm_blocks, TPB, 0, stream>>>(x, 512, W0, t, N, 512);
    k_selfloop<<<gridNF, TPB, 0, stream>>>(t, dinv, agg, nf);
    k_scatter<<<gridScat, TPB, 0, stream>>>(src, dst, dinv, t, agg, E);
    k_bias_relu_emb<<<gridNF, TPB, 0, stream>>>(agg, b0, emb, 0, nf);

    // --- GCN layer 1: emb[:,0:128] @ W1 ---
    k_gemm_f128<<<gemm_blocks, TPB, 0, stream>>>(emb, 384, W1, t, N, 128);
    k_selfloop<<<gridNF, TPB, 0, stream>>>(t, dinv, agg, nf);
    k_scatter<<<gridScat, TPB, 0, stream>>>(src, dst, dinv, t, agg, E);
    k_bias_relu_emb<<<gridNF, TPB, 0, stream>>>(agg, b1, emb, 128, nf);

    // --- GCN layer 2: emb[:,128:256] @ W2 ---
    k_gemm_f128<<<gemm_blocks, TPB, 0, stream>>>(emb + 128, 384, W2, t, N, 128);
    k_selfloop<<<gridNF, TPB, 0, stream>>>(t, dinv, agg, nf);
    k_scatter<<<gridScat, TPB, 0, stream>>>(src, dst, dinv, t, agg, E);
    k_bias_relu_emb<<<gridNF, TPB, 0, stream>>>(agg, b2, emb, 256, nf);

    // --- MLP: t = elu(emb[N,384] @ Wm0 + bm0) ---
    k_gemm_f128<<<gemm_blocks, TPB, 0, stream>>>(emb, 384, Wm0, t, N, 384);
    k_bias_elu<<<gridNF, TPB, 0, stream>>>(t, bm0, nf);

    // --- logits + softmax ---
    k_head<<<(N + TPB - 1) / TPB, TPB, 0, stream>>>(t, Wm1, bm1, logits, probs, N);
}